// SAILoss_10857677324423
// MI455X (gfx1250) — compile-verified
//
#include <hip/hip_runtime.h>

// ---------------------------------------------------------------------------
// SAI loss pipeline for MI455X (gfx1250, wave32).
//   K0  prep      : wm = w*mask, u = 0, zero scalar accumulators
//   K1  norm      : sum|A_diag| + sum|A_off|, sum(mask)        -> ws[0], ws[1]
//   K2  scatter   : u = G^T wm  (global f32 atomics) + pack neighbors to i32
//   K3  gatherG   : v = G u + eps*wm        (WMMA f32 16x16x4 diagonal trick)
//   K4  gatherA   : y = A v; d=(y/norm - wm)*mask; sum d^2      -> ws[2]
//   K5  final     : out = ws[2] / (ws[1] + 1e-6)
// Memory-bound: ~300 MB of traffic vs ~80 MFLOP -> optimize bytes + gather
// latency, keep everything f32 (WMMA f32 16x16x4 is full precision).
// ---------------------------------------------------------------------------

typedef __attribute__((ext_vector_type(2))) float v2f;
typedef __attribute__((ext_vector_type(8))) float v8f;

#define BB   2
#define NN   262144
#define BN   (BB * NN)
#define ST   24
#define EPSF 1e-4f

// ----------------------------------------------------- async global->LDS path
#if defined(__gfx1250__) && __has_builtin(__builtin_amdgcn_global_load_async_to_lds_b32) && \
    __has_builtin(__builtin_amdgcn_global_load_async_to_lds_b128)
#define ASYNC_LDS 1
#else
#define ASYNC_LDS 0
#endif

#if ASYNC_LDS
// Builtin params are typed, address-space-qualified pointers:
//   b128: (AS1 v4i*, AS3 v4i*, Imm offset, Imm cpol)
//   b32 : (AS1 int*, AS3 int*, Imm offset, Imm cpol)
typedef int v4i __attribute__((__vector_size__(16)));
typedef __attribute__((address_space(1))) v4i as1_v4i;
typedef __attribute__((address_space(3))) v4i as3_v4i;
typedef __attribute__((address_space(1))) int as1_int;
typedef __attribute__((address_space(3))) int as3_int;
__device__ __forceinline__ as1_v4i* g128(const void* p) {
  return (as1_v4i*)(unsigned long long)p;                 // flat == global address
}
__device__ __forceinline__ as3_v4i* l128(void* p) {
  return (as3_v4i*)(unsigned int)(unsigned long long)p;   // low 32 bits = LDS offset
}
__device__ __forceinline__ as1_int* g32(const void* p) {
  return (as1_int*)(unsigned long long)p;
}
__device__ __forceinline__ as3_int* l32(void* p) {
  return (as3_int*)(unsigned int)(unsigned long long)p;
}
#endif

__device__ __forceinline__ void async_wait0() {
#if ASYNC_LDS
#if __has_builtin(__builtin_amdgcn_s_wait_asynccnt)
  __builtin_amdgcn_s_wait_asynccnt(0);
#else
  asm volatile("s_wait_asynccnt 0x0" ::: "memory");
#endif
#endif
}

// ---------------------------------------------------------------- K0: prep
__global__ __launch_bounds__(256) void k_prep(const float* __restrict__ w,
                                              const float* __restrict__ mask,
                                              float* __restrict__ wm,
                                              float* __restrict__ u,
                                              float* __restrict__ scal) {
  int i = blockIdx.x * 256 + threadIdx.x;
  if (i < 4) scal[i] = 0.0f;
  if (i < BN) {
    wm[i] = w[i] * mask[i];
    u[i]  = 0.0f;
  }
}

// ---------------------------------------------------------------- K1: norm
__global__ __launch_bounds__(256) void k_norm(const float* __restrict__ Ad,
                                              const float* __restrict__ Ao,
                                              const float* __restrict__ mask,
                                              float* __restrict__ scal) {
  __shared__ float s1[256];
  __shared__ float s2[256];
  float a = 0.0f, act = 0.0f;
  const long total  = (long)BN * ST;
  const long stride = (long)gridDim.x * 256;
  for (long t = (long)blockIdx.x * 256 + threadIdx.x; t < total; t += stride)
    a += fabsf(Ao[t]);
  for (int t = blockIdx.x * 256 + threadIdx.x; t < BN; t += gridDim.x * 256) {
    a   += fabsf(Ad[t]);
    act += mask[t];
  }
  s1[threadIdx.x] = a;
  s2[threadIdx.x] = act;
  __syncthreads();
  for (int o = 128; o > 0; o >>= 1) {
    if (threadIdx.x < o) {
      s1[threadIdx.x] += s1[threadIdx.x + o];
      s2[threadIdx.x] += s2[threadIdx.x + o];
    }
    __syncthreads();
  }
  if (threadIdx.x == 0) {
    atomicAdd(&scal[0], s1[0]);
    atomicAdd(&scal[1], s2[0]);
  }
}

// ------------------------------------------------------------- K2: scatter
// u = G^T wm via global f32 atomics (target is 2 MB -> resolves in L2).
// Also packs int64 neighbor ids to int32 (halves neighbor bytes for K3/K4).
__global__ __launch_bounds__(256) void k_scatter(const float* __restrict__ G,
                                                 const long long* __restrict__ nbr,
                                                 const float* __restrict__ wm,
                                                 float* __restrict__ u,
                                                 int* __restrict__ p32,
                                                 int doPack) {
  __shared__ __align__(16) float cf[256 * 25];  // 25.6 KB coeff stage
  const int f0 = blockIdx.x * 256;
#if ASYNC_LDS
  for (int t = threadIdx.x * 4; t < 256 * 25; t += 256 * 4)
    __builtin_amdgcn_global_load_async_to_lds_b128(
        g128(G + (size_t)f0 * 25 + t), l128(&cf[t]), 0, 0);
  async_wait0();
#else
  for (int t = threadIdx.x; t < 256 * 25; t += 256)
    cf[t] = G[(size_t)f0 * 25 + t];
#endif
  __syncthreads();

  const int i = f0 + threadIdx.x;
  const int b = i / NN;
  const float wi = wm[i];
  const float* c = &cf[threadIdx.x * 25];
  const int* nb = (const int*)(nbr + (size_t)i * ST);  // low dwords (ids < 2^31)
  float* ub = u + (size_t)b * NN;

  int id[ST];
#pragma unroll
  for (int j = 0; j < ST; ++j) id[j] = nb[2 * j];
  if (doPack) {
#pragma unroll
    for (int j = 0; j < ST; ++j) p32[(size_t)i * ST + j] = id[j];
  }
  atomicAdd(&u[i], c[0] * wi);
#pragma unroll
  for (int j = 0; j < ST; ++j)
    if (id[j] < NN) atomicAdd(&ub[id[j]], c[j + 1] * wi);
}

// --------------------------------------------------- WMMA gather core (K3/K4)
// Wave handles 16 nodes. diag(A x B) = 16 stencil dot products in full f32,
// accumulated over 7 chunks of K=4 (25 slots padded to 28).
// A 16x4 f32 layout (ISA 7.12.2): lanes 0-15 -> K=kk,kk+1 ; lanes 16-31 ->
// K=kk+2,kk+3 (same M per lane%16). B mirrored with N = lane%16.
// All gathers are branchless: clamped-index loads + v_cndmask selects, so the
// compiler batches the id loads and value loads instead of per-load waits.
__device__ __forceinline__ v8f wmma_stencil_tile(const float* __restrict__ cc,  // 25 coeffs (LDS)
                                                 const int* __restrict__ nbRow, // neighbor ids
                                                 int nbStride,                  // 1 packed / 2 i64
                                                 const float* __restrict__ xb,  // batch-local x
                                                 int selfLocal, int half) {
  float av[14], xv[14];
  int   id[14];
  // ---- all coefficient reads (LDS, unconditional at clamped slot)
#pragma unroll
  for (int it = 0; it < 7; ++it) {
    const int s0 = it * 4 + 2 * half, s1 = s0 + 1;
    const int sc0 = s0 > 24 ? 24 : s0, sc1 = s1 > 24 ? 24 : s1;
    float c0 = cc[sc0], c1 = cc[sc1];
    av[2 * it]     = (s0 < 25) ? c0 : 0.0f;
    av[2 * it + 1] = (s1 < 25) ? c1 : 0.0f;
  }
  // ---- all neighbor-id loads (unconditional, clamped j)
#pragma unroll
  for (int it = 0; it < 7; ++it) {
    const int s0 = it * 4 + 2 * half, s1 = s0 + 1;
    int j0 = s0 - 1; j0 = j0 < 0 ? 0 : (j0 > 23 ? 23 : j0);
    int j1 = s1 - 1; j1 = j1 > 23 ? 23 : j1;
    id[2 * it]     = nbRow[j0 * nbStride];
    id[2 * it + 1] = nbRow[j1 * nbStride];
  }
  // ---- all x gathers (unconditional from safe address, then select)
#pragma unroll
  for (int it = 0; it < 7; ++it) {
    const int s0 = it * 4 + 2 * half, s1 = s0 + 1;
    int i0 = (s0 == 0) ? selfLocal : id[2 * it];
    int i1 = id[2 * it + 1];
    bool v0 = (s0 == 0) || ((s0 < 25) && (i0 < NN));
    bool v1 = (s1 < 25) && (i1 < NN);
    float x0 = xb[v0 ? i0 : 0];
    float x1 = xb[v1 ? i1 : 0];
    xv[2 * it]     = v0 ? x0 : 0.0f;
    xv[2 * it + 1] = v1 ? x1 : 0.0f;
  }
  // ---- 7 back-to-back WMMA accumulates (full f32)
  v8f acc = {};
#pragma unroll
  for (int it = 0; it < 7; ++it) {
    v2f a, bv;
    a.x  = av[2 * it]; a.y  = av[2 * it + 1];
    bv.x = xv[2 * it]; bv.y = xv[2 * it + 1];
    acc = __builtin_amdgcn_wmma_f32_16x16x4_f32(false, a, false, bv,
                                                (short)0, acc, false, false);
  }
  return acc;
}

// ------------------------------------------------------- K3: v = G u + eps*wm
__global__ __launch_bounds__(256) void k_gatherG(const float* __restrict__ G,
                                                 const int* __restrict__ nbi,
                                                 int nbStride,
                                                 const float* __restrict__ x,  // u
                                                 const float* __restrict__ wm,
                                                 float* __restrict__ out) {    // v
  __shared__ __align__(16) float cf[128 * 25];  // 12.8 KB coeff stage
  __shared__ float sd[8][32][9];                // diag extract (stride 9: conflict-free)
  const int tid = threadIdx.x;
  const int wave = tid >> 5, lane = tid & 31;
  const int half = lane >> 4, l16 = lane & 15;
  const int f0 = blockIdx.x * 128;

#if ASYNC_LDS
  for (int t = tid * 4; t < 128 * 25; t += 256 * 4)
    __builtin_amdgcn_global_load_async_to_lds_b128(
        g128(G + (size_t)f0 * 25 + t), l128(&cf[t]), 0, 0);
  async_wait0();
#else
  for (int t = tid; t < 128 * 25; t += 256)
    cf[t] = G[(size_t)f0 * 25 + t];
#endif
  __syncthreads();

  const int fw = f0 + wave * 16;     // 16-node tile (128 | N: never straddles a batch)
  const int b  = fw / NN;
  const float* xb = x + (size_t)b * NN;
  const int f = fw + l16;
  const int* nbRow = nbi + (size_t)f * ST * nbStride;
  const float* cc = &cf[(wave * 16 + l16) * 25];

  v8f acc = wmma_stencil_tile(cc, nbRow, nbStride, xb, f - b * NN, half);

#pragma unroll
  for (int c = 0; c < 8; ++c) sd[wave][lane][c] = acc[c];
  __syncthreads();
  if (lane < 16) {
    float dg = (l16 < 8) ? sd[wave][l16][l16] : sd[wave][16 + l16][l16 - 8];
    out[f] = dg + EPSF * wm[f];
  }
}

// ------------------------------------------------------------- K4: loss pass
__global__ __launch_bounds__(256) void k_gatherA(const float* __restrict__ Ad,
                                                 const float* __restrict__ Ao,
                                                 const int* __restrict__ nbi,
                                                 int nbStride,
                                                 const float* __restrict__ x,  // v
                                                 const float* __restrict__ wm,
                                                 const float* __restrict__ mask,
                                                 float* __restrict__ scal) {
  __shared__ __align__(16) float cf[128 * 25];
  __shared__ float sd[8][32][9];
  __shared__ float rs[256];
  const int tid = threadIdx.x;
  const int wave = tid >> 5, lane = tid & 31;
  const int half = lane >> 4, l16 = lane & 15;
  const int f0 = blockIdx.x * 128;

#if ASYNC_LDS
  for (int t = tid; t < 128 * ST; t += 256) {
    int node = t / ST, j = t % ST;
    __builtin_amdgcn_global_load_async_to_lds_b32(
        g32(Ao + (size_t)f0 * ST + t), l32(&cf[node * 25 + 1 + j]), 0, 0);
  }
  for (int t = tid; t < 128; t += 256)
    __builtin_amdgcn_global_load_async_to_lds_b32(
        g32(Ad + f0 + t), l32(&cf[t * 25]), 0, 0);
  async_wait0();
#else
  for (int t = tid; t < 128 * ST; t += 256) {
    int node = t / ST, j = t % ST;
    cf[node * 25 + 1 + j] = Ao[(size_t)f0 * ST + t];
  }
  for (int t = tid; t < 128; t += 256)
    cf[t * 25] = Ad[f0 + t];
#endif
  __syncthreads();

  const float na   = scal[0] / (scal[1] * 25.0f + 1e-6f);
  const float invn = 1.0f / (na + 1e-8f);

  const int fw = f0 + wave * 16;
  const int b  = fw / NN;
  const float* xb = x + (size_t)b * NN;
  const int f = fw + l16;
  const int* nbRow = nbi + (size_t)f * ST * nbStride;
  const float* cc = &cf[(wave * 16 + l16) * 25];

  v8f acc = wmma_stencil_tile(cc, nbRow, nbStride, xb, f - b * NN, half);

#pragma unroll
  for (int c = 0; c < 8; ++c) sd[wave][lane][c] = acc[c];
  __syncthreads();

  float contrib = 0.0f;
  if (lane < 16) {
    float dg = (l16 < 8) ? sd[wave][l16][l16] : sd[wave][16 + l16][l16 - 8];
    float z  = dg * invn;
    float d  = (z - wm[f]) * mask[f];
    contrib  = d * d;
  }
  rs[tid] = contrib;
  __syncthreads();
  for (int o = 128; o > 0; o >>= 1) {
    if (tid < o) rs[tid] += rs[tid + o];
    __syncthreads();
  }
  if (tid == 0) atomicAdd(&scal[2], rs[0]);
}

// ------------------------------------------------------------- K5: finalize
__global__ void k_final(const float* __restrict__ scal, float* __restrict__ out) {
  out[0] = scal[2] / (scal[1] + 1e-6f);
}

// ---------------------------------------------------------------------------
extern "C" void kernel_launch(void* const* d_in, const int* in_sizes, int n_in,
                              void* d_out, int out_size, void* d_ws, size_t ws_size,
                              hipStream_t stream) {
  const float*     G   = (const float*)d_in[0];     // [B,N,25]
  const float*     Ad  = (const float*)d_in[1];     // [B,N,1]
  const float*     Ao  = (const float*)d_in[2];     // [B,N,24]
  const float*     w   = (const float*)d_in[3];     // [B,N,1]
  const float*     mk  = (const float*)d_in[4];     // [B,N,1]
  const long long* nbr = (const long long*)d_in[5]; // [B,N,24] int64
  float* out = (float*)d_out;

  float* wsf  = (float*)d_ws;
  float* scal = wsf;            // [0]=|A| sum, [1]=active, [2]=diff^2 sum
  float* wm   = wsf + 64;       // [B*N]
  float* u    = wm + BN;        // [B*N]
  float* v    = u + BN;         // [B*N]
  int*   p32  = (int*)(v + BN); // optional [B*N*24] packed int32 neighbor ids

  const size_t needBase = (size_t)(64 + 3 * BN) * sizeof(float);
  const size_t needPack = needBase + (size_t)BN * ST * sizeof(int);
  const int doPack = (ws_size >= needPack) ? 1 : 0;   // fixed per run -> deterministic

  const int* nbi   = doPack ? (const int*)p32 : (const int*)nbr;
  const int  nbStr = doPack ? 1 : 2;                  // i32 packed vs i64 low dword

  k_prep   <<<BN / 256, 256, 0, stream>>>(w, mk, wm, u, scal);
  k_norm   <<<1024,     256, 0, stream>>>(Ad, Ao, mk, scal);
  k_scatter<<<BN / 256, 256, 0, stream>>>(G, nbr, wm, u, p32, doPack);
  k_gatherG<<<BN / 128, 256, 0, stream>>>(G, nbi, nbStr, u, wm, v);
  k_gatherA<<<BN / 128, 256, 0, stream>>>(Ad, Ao, nbi, nbStr, v, wm, mk, scal);
  k_final  <<<1, 1, 0, stream>>>(scal, out);
  (void)in_sizes; (void)n_in; (void)out_size; (void)ws_size;
}